// TensCoinMiner_18116172054737
// MI455X (gfx1250) — compile-verified
//
#include <hip/hip_runtime.h>

typedef __attribute__((ext_vector_type(2))) float v2f;
typedef __attribute__((ext_vector_type(8))) float v8f;

#define HIDDEN  256
#define ROUNDS  64
#define BTILE   64        // batch rows per workgroup
#define SSTRIDE 260       // 256 + 4 floats pad -> conflict-free strided LDS reads

__global__ __launch_bounds__(256) void tenscoin_wmma_kernel(
    const int*   __restrict__ nonce,     // [B, 32]  bytes 0..255
    const float* __restrict__ noise,     // [B, 256]
    const float* __restrict__ matrices,  // [64, 256, 256]
    float*       __restrict__ out)       // [B, 32]  packed bytes as float
{
    __shared__ __align__(16) float Sbuf[BTILE * SSTRIDE];

    const int tid    = threadIdx.x;
    const int wave   = tid >> 5;          // wave32
    const int lane   = tid & 31;
    const int lane16 = lane & 15;
    const int hi     = lane >> 4;         // 0: K pair {0,1}; 1: K pair {2,3}
    const int b0     = blockIdx.x * BTILE;

    // ---- init state from nonce bits (bit j of byte, LSB-first) ----
    for (int idx = tid; idx < BTILE * HIDDEN; idx += 256) {
        int row  = idx >> 8;
        int col  = idx & 255;
        int byte = nonce[(b0 + row) * 32 + (col >> 3)];
        Sbuf[row * SSTRIDE + col] = (float)((byte >> (col & 7)) & 1);
    }
    __syncthreads();

    // ---- per-wave tile assignment: col tiles {2w, 2w+1}, row tiles 0..3 ----
    const int ctbase = wave * 32;

    // noise tiles in C/D layout: VGPR v -> M = v + 8*hi ; N = lane16
    v8f nz[4][2];
    #pragma unroll
    for (int rt = 0; rt < 4; ++rt)
        #pragma unroll
        for (int ct = 0; ct < 2; ++ct) {
            int col = ctbase + ct * 16 + lane16;
            #pragma unroll
            for (int v = 0; v < 8; ++v) {
                int row = b0 + rt * 16 + v + hi * 8;
                nz[rt][ct][v] = noise[row * HIDDEN + col];
            }
        }

    // B[k][n] = M[n][k]: lane supplies n = ctbase + ct*16 + lane16,
    // reading consecutive K pair at offset 2*hi (ISA B layout: lanes 0-15
    // hold K=0,1 ; lanes 16-31 hold K=2,3).
    const size_t mrow0 = (size_t)(ctbase + 0 * 16 + lane16) * HIDDEN + 2 * hi;
    const size_t mrow1 = (size_t)(ctbase + 1 * 16 + lane16) * HIDDEN + 2 * hi;

    for (int r = 0; r < ROUNDS; ++r) {
        const float* __restrict__ mat = matrices + (size_t)r * (HIDDEN * HIDDEN);

        // warm next round's matrix rows into caches (global_prefetch_b8)
        if (r + 1 < ROUNDS) {
            const float* nm = mat + HIDDEN * HIDDEN;
            __builtin_prefetch(&nm[mrow0], 0, 1);
            __builtin_prefetch(&nm[mrow1], 0, 1);
        }

        v8f acc[4][2];
        #pragma unroll
        for (int rt = 0; rt < 4; ++rt) {
            acc[rt][0] = nz[rt][0];
            acc[rt][1] = nz[rt][1];
        }

        #pragma unroll 4
        for (int k = 0; k < HIDDEN; k += 4) {
            // A: state rows M = lane16 of row-tile rt, K pair at k + 2*hi (LDS)
            v2f a[4];
            #pragma unroll
            for (int rt = 0; rt < 4; ++rt)
                a[rt] = *(const v2f*)&Sbuf[(rt * 16 + lane16) * SSTRIDE + k + 2 * hi];

            // B: matrix rows (= output cols), K pair at k + 2*hi (global -> L2 hit)
            v2f bv0 = *(const v2f*)&mat[mrow0 + k];
            v2f bv1 = *(const v2f*)&mat[mrow1 + k];

            #pragma unroll
            for (int rt = 0; rt < 4; ++rt) {
                acc[rt][0] = __builtin_amdgcn_wmma_f32_16x16x4_f32(
                    false, a[rt], false, bv0, (short)0, acc[rt][0], false, false);
                acc[rt][1] = __builtin_amdgcn_wmma_f32_16x16x4_f32(
                    false, a[rt], false, bv1, (short)0, acc[rt][1], false, false);
            }
        }

        __syncthreads();   // all waves done reading old state

        // fmod(x, 2) with C sign semantics, exact in fp32, then write back
        #pragma unroll
        for (int rt = 0; rt < 4; ++rt)
            #pragma unroll
            for (int ct = 0; ct < 2; ++ct) {
                int col = ctbase + ct * 16 + lane16;
                #pragma unroll
                for (int v = 0; v < 8; ++v) {
                    float x = acc[rt][ct][v];
                    x = x - 2.0f * truncf(x * 0.5f);
                    Sbuf[(rt * 16 + v + hi * 8) * SSTRIDE + col] = x;
                }
            }

        __syncthreads();   // new state visible to all waves
    }

    // ---- pack bits: bit j set iff |state| >= 1.0, LSB-first per byte ----
    for (int idx = tid; idx < BTILE * 32; idx += 256) {
        int row  = idx >> 5;
        int byte = idx & 31;
        int accb = 0;
        #pragma unroll
        for (int j = 0; j < 8; ++j) {
            float s = Sbuf[row * SSTRIDE + byte * 8 + j];
            if (__builtin_fabsf(s) >= 1.0f) accb |= (1 << j);
        }
        out[(size_t)(b0 + row) * 32 + byte] = (float)accb;
    }
}

extern "C" void kernel_launch(void* const* d_in, const int* in_sizes, int n_in,
                              void* d_out, int out_size, void* d_ws, size_t ws_size,
                              hipStream_t stream) {
    const int*   nonce    = (const int*)d_in[0];    // [B,32] int32
    const float* noise    = (const float*)d_in[1];  // [B,256] fp32
    const float* matrices = (const float*)d_in[2];  // [64,256,256] fp32
    float*       out      = (float*)d_out;          // [B,32]

    const int B    = in_sizes[0] / 32;   // 65536
    const int grid = B / BTILE;          // 1024 workgroups

    hipLaunchKernelGGL(tenscoin_wmma_kernel, dim3(grid), dim3(256), 0, stream,
                       nonce, noise, matrices, out);
}